// FeedForwardNN_49632642072955
// MI455X (gfx1250) — compile-verified
//
#include <hip/hip_runtime.h>

typedef __attribute__((ext_vector_type(2))) float v2f;
typedef __attribute__((ext_vector_type(8))) float v8f;
typedef long long ll;

// ---------------------------------------------------------------------------
// Elementwise / edge kernels
// ---------------------------------------------------------------------------
__global__ void k_zero(float* __restrict__ p, ll n) {
  ll t = (ll)blockIdx.x * blockDim.x + threadIdx.x;
  if (t < n) p[t] = 0.0f;
}

// deg[dst] += w  (deg pre-zeroed)
__global__ void k_deg(const ll* __restrict__ ei, const float* __restrict__ ew,
                      float* __restrict__ deg, ll E) {
  ll t = (ll)blockIdx.x * blockDim.x + threadIdx.x;
  if (t >= E) return;
  ll d = ei[E + t];
  atomicAdd(&deg[d], ew[t]);
}

// in-place deg -> d^{-1/2}
__global__ void k_dinv(float* __restrict__ deg, ll n) {
  ll t = (ll)blockIdx.x * blockDim.x + threadIdx.x;
  if (t >= n) return;
  float v = deg[t];
  deg[t] = (v > 0.0f) ? rsqrtf(fmaxf(v, 1e-12f)) : 0.0f;
}

// norm[e] = -dinv[dst] * w * dinv[src]
__global__ void k_norm(const ll* __restrict__ ei, const float* __restrict__ ew,
                       const float* __restrict__ dinv, float* __restrict__ nrm, ll E) {
  ll t = (ll)blockIdx.x * blockDim.x + threadIdx.x;
  if (t >= E) return;
  ll s = ei[t];
  ll d = ei[E + t];
  nrm[t] = -dinv[d] * ew[t] * dinv[s];
}

// out[dst,:] += norm[e] * h[src,:]   (out pre-zeroed). One thread = one
// (edge, 4-feature chunk); a wave covers one edge's features contiguously.
__global__ void k_scatter(const float* __restrict__ h, const ll* __restrict__ ei,
                          const float* __restrict__ nrm, float* __restrict__ out,
                          ll E, int F, int cs /* log2(F/4) */) {
  ll t = (ll)blockIdx.x * blockDim.x + threadIdx.x;
  ll total = E << cs;
  if (t >= total) return;
  ll e = t >> cs;
  int c = (int)(t & ((1 << cs) - 1)) << 2;
  float w = nrm[e];
  ll s = ei[e];
  ll d = ei[E + e];
  float4 v = *(const float4*)(h + s * (ll)F + c);
  float* o = out + d * (ll)F + c;
  atomicAdd(o + 0, w * v.x);
  atomicAdd(o + 1, w * v.y);
  atomicAdd(o + 2, w * v.z);
  atomicAdd(o + 3, w * v.w);
}

// Tx2 = 2*prop(Tx1) - Tx0, in place on t2 (t2 holds prop(Tx1) on entry)
__global__ void k_combine(float* __restrict__ t2, const float* __restrict__ t0, ll n) {
  ll t = (ll)blockIdx.x * blockDim.x + threadIdx.x;
  if (t < n) t2[t] = 2.0f * t2[t] - t0[t];
}

// ---------------------------------------------------------------------------
// Fused Chebyshev GEMM:  out = relu?( bias + T0@W[0] + T1@W[1] + T2@W[2] )
//
// Block = 128 threads = 4 waves. Each wave owns a 16-row M tile and computes
// the full 16 x (NT*16) output strip: one fp32 A fragment feeds NT WMMAs.
//
// W is staged through LDS in 16-deep k-blocks in a FRAGMENT-READY layout:
//   sW[t][kpair][col][2] = { W[t][kb+2*kpair][col], W[t][kb+2*kpair+1][col] }
// so each lane's B fragment is a single contiguous ds_load_b64 straight into
// an even-aligned VGPR pair (no repack moves before v_wmma).
//
// Requires: Fout = NT*16, Fin % 16 == 0, N % 16 == 0. Waves past the last
// M tile clamp to it (duplicate identical-value stores), keeping EXEC
// all-ones and barrier participation uniform, as WMMA requires.
// ---------------------------------------------------------------------------
template <int NT>
__global__ __launch_bounds__(128) void k_cheb_gemm(
    const float* __restrict__ T0, const float* __restrict__ T1,
    const float* __restrict__ T2, const float* __restrict__ W,
    const float* __restrict__ bias, float* __restrict__ out,
    int Fin, int mtiles, int doRelu) {
  constexpr int FOUT = NT * 16;
  // [3][8 kpairs][FOUT cols][2] floats = 3*16*FOUT floats (<= 24 KB)
  __shared__ float sW[3 * 8 * FOUT * 2];

  const int lane = threadIdx.x & 31;
  const int wave = threadIdx.x >> 5;
  int mt = blockIdx.x * 4 + wave;
  if (mt >= mtiles) mt = mtiles - 1;  // clamp (see note above)
  const int m0 = mt << 4;

  const float* Tm[3] = {T0, T1, T2};

  v8f acc[NT];
#pragma unroll
  for (int nt = 0; nt < NT; ++nt) {
    const float bv = bias[nt * 16 + (lane & 15)];
#pragma unroll
    for (int r = 0; r < 8; ++r) acc[nt][r] = bv;
  }

  for (int kb = 0; kb < Fin; kb += 16) {
    __syncthreads();  // previous k-block's LDS reads done before overwrite
    // Cooperative stage with k-pair interleave. Per pair-element p:
    //   t = matrix, kp = k-pair within block, c = column.
    // Two coalesced global b32 reads + one contiguous ds_store_b64.
    for (int p = threadIdx.x; p < 3 * 8 * FOUT; p += 128) {
      const int t = p / (8 * FOUT);
      const int rem = p - t * 8 * FOUT;
      const int kp = rem / FOUT;
      const int c = rem - kp * FOUT;
      const float* g = W + ((ll)t * Fin + kb + 2 * kp) * FOUT + c;
      v2f v;
      v.x = g[0];
      v.y = g[FOUT];
      *(v2f*)(sW + 2 * p) = v;
    }
    __syncthreads();

#pragma unroll
    for (int t = 0; t < 3; ++t) {
      const float* __restrict__ Tt = Tm[t];
#pragma unroll
      for (int kk = 0; kk < 16; kk += 4) {
        // A frag (16x4 fp32): lanes 0-15 -> (K0,K1); 16-31 -> (K2,K3)
        const float* pa = Tt + (ll)(m0 + (lane & 15)) * Fin +
                          (kb + kk + ((lane >> 4) << 1));
        v2f a;
        a.x = pa[0];
        a.y = pa[1];
        // B frags: one ds_load_b64 each, fragment-ready layout
        const int kp = (kk >> 1) + (lane >> 4);
        const float* sB = sW + 2 * ((t * 8 + kp) * FOUT + (lane & 15));
#pragma unroll
        for (int nt = 0; nt < NT; ++nt) {
          v2f b = *(const v2f*)(sB + 2 * (nt * 16));
          acc[nt] = __builtin_amdgcn_wmma_f32_16x16x4_f32(
              false, a, false, b, (short)0, acc[nt], false, false);
        }
      }
    }
  }

  // C/D layout: VGPR r -> row m0 + r + 8*(lane>=16), col n0 + (lane&15)
  const int row0 = m0 + ((lane >> 4) << 3);
#pragma unroll
  for (int nt = 0; nt < NT; ++nt) {
    const int col = nt * 16 + (lane & 15);
#pragma unroll
    for (int r = 0; r < 8; ++r) {
      float v = acc[nt][r];
      if (doRelu) v = fmaxf(v, 0.0f);
      out[(ll)(row0 + r) * FOUT + col] = v;
    }
  }
}

// ---------------------------------------------------------------------------
// Layer-3 padding (Fout 40 -> 48) and trim
// ---------------------------------------------------------------------------
__global__ void k_pad_w3(const float* __restrict__ W3, float* __restrict__ W3p) {
  int t = blockIdx.x * blockDim.x + threadIdx.x;  // 3*64*48
  if (t >= 3 * 64 * 48) return;
  int col = t % 48;
  int row = t / 48;
  W3p[t] = (col < 40) ? W3[row * 40 + col] : 0.0f;
}
__global__ void k_pad_b3(const float* __restrict__ b3, float* __restrict__ b3p) {
  int t = threadIdx.x;  // 48 threads
  if (t < 48) b3p[t] = (t < 40) ? b3[t] : 0.0f;
}
__global__ void k_trim(const float* __restrict__ acc48, float* __restrict__ out, ll total) {
  ll t = (ll)blockIdx.x * blockDim.x + threadIdx.x;  // total = N*40
  if (t >= total) return;
  ll n = t / 40;
  int c = (int)(t % 40);
  out[t] = acc48[n * 48 + c];
}

// ---------------------------------------------------------------------------
// Launcher
// ---------------------------------------------------------------------------
static inline int nblk(ll n, int b) { return (int)((n + b - 1) / b); }

extern "C" void kernel_launch(void* const* d_in, const int* in_sizes, int n_in,
                              void* d_out, int out_size, void* d_ws, size_t ws_size,
                              hipStream_t stream) {
  const float* x  = (const float*)d_in[0];   // [N,128]
  const ll*    ei = (const ll*)d_in[1];      // [2,E] int64
  const float* ew = (const float*)d_in[2];   // [E]
  const float* W1 = (const float*)d_in[3];   // [3,128,128]
  const float* b1 = (const float*)d_in[4];
  const float* W2 = (const float*)d_in[5];   // [3,128,64]
  const float* b2 = (const float*)d_in[6];
  const float* W3 = (const float*)d_in[7];   // [3,64,40]
  const float* b3 = (const float*)d_in[8];
  float* out = (float*)d_out;

  const ll N = in_sizes[0] / 128;  // 50000 (multiple of 16)
  const ll E = in_sizes[2];        // 800000
  const int mtiles = (int)(N / 16);
  const int gemmBlocks = (mtiles + 3) / 4;

  // Workspace layout (floats)
  float* ws   = (float*)d_ws;
  float* nrm  = ws;                      // E
  float* deg  = nrm + E;                 // N  (becomes dinv in place)
  float* tx1  = deg + N;                 // N*128
  float* tx2  = tx1 + N * 128;           // N*128
  float* h1   = tx2 + N * 128;           // N*128
  float* h2   = h1 + N * 128;            // N*64
  float* a48  = h2 + N * 64;             // N*48
  float* w3p  = a48 + N * 48;            // 3*64*48
  float* b3p  = w3p + 3 * 64 * 48;       // 48

  const int B = 256;

  // ---- edge normalization ----
  k_zero<<<nblk(N, B), B, 0, stream>>>(deg, N);
  k_deg<<<nblk(E, B), B, 0, stream>>>(ei, ew, deg, E);
  k_dinv<<<nblk(N, B), B, 0, stream>>>(deg, N);
  k_norm<<<nblk(E, B), B, 0, stream>>>(ei, ew, deg, nrm, E);

  // ---- layer 1: 128 -> 128, relu ----
  k_zero<<<nblk(N * 128, B), B, 0, stream>>>(tx1, N * 128);
  k_scatter<<<nblk(E << 5, B), B, 0, stream>>>(x, ei, nrm, tx1, E, 128, 5);
  k_zero<<<nblk(N * 128, B), B, 0, stream>>>(tx2, N * 128);
  k_scatter<<<nblk(E << 5, B), B, 0, stream>>>(tx1, ei, nrm, tx2, E, 128, 5);
  k_combine<<<nblk(N * 128, B), B, 0, stream>>>(tx2, x, N * 128);
  k_cheb_gemm<8><<<gemmBlocks, 128, 0, stream>>>(x, tx1, tx2, W1, b1, h1,
                                                 128, mtiles, 1);

  // ---- layer 2: 128 -> 64, relu ----
  k_zero<<<nblk(N * 128, B), B, 0, stream>>>(tx1, N * 128);
  k_scatter<<<nblk(E << 5, B), B, 0, stream>>>(h1, ei, nrm, tx1, E, 128, 5);
  k_zero<<<nblk(N * 128, B), B, 0, stream>>>(tx2, N * 128);
  k_scatter<<<nblk(E << 5, B), B, 0, stream>>>(tx1, ei, nrm, tx2, E, 128, 5);
  k_combine<<<nblk(N * 128, B), B, 0, stream>>>(tx2, h1, N * 128);
  k_cheb_gemm<4><<<gemmBlocks, 128, 0, stream>>>(h1, tx1, tx2, W2, b2, h2,
                                                 128, mtiles, 1);

  // ---- layer 3: 64 -> 40 (padded to 48), no relu ----
  k_pad_w3<<<nblk(3 * 64 * 48, B), B, 0, stream>>>(W3, w3p);
  k_pad_b3<<<1, 48, 0, stream>>>(b3, b3p);
  k_zero<<<nblk(N * 64, B), B, 0, stream>>>(tx1, N * 64);
  k_scatter<<<nblk(E << 4, B), B, 0, stream>>>(h2, ei, nrm, tx1, E, 64, 4);
  k_zero<<<nblk(N * 64, B), B, 0, stream>>>(tx2, N * 64);
  k_scatter<<<nblk(E << 4, B), B, 0, stream>>>(tx1, ei, nrm, tx2, E, 64, 4);
  k_combine<<<nblk(N * 64, B), B, 0, stream>>>(tx2, h2, N * 64);
  k_cheb_gemm<3><<<gemmBlocks, 128, 0, stream>>>(h2, tx1, tx2, w3p, b3p, a48,
                                                 64, mtiles, 0);
  k_trim<<<nblk(N * 40, B), B, 0, stream>>>(a48, out, N * 40);
}